// RWKV_TimeMix_17111149707627
// MI455X (gfx1250) — compile-verified
//
#include <hip/hip_runtime.h>
#include <hip/hip_bf16.h>
#include <math.h>

// ---------------------------------------------------------------------------
// RWKV TimeMix forward for MI455X (gfx1250, wave32, WMMA + async-LDS DMA).
//
// Shapes: B=4, T=1024, HID=1024, N_ATTN=1024, H=64, HS=16, CTX=1024.
// Memory-bound (~43 GFLOP vs ~150MB essential HBM traffic @ 23.3 TB/s) =>
// exact-precision fp32 matrix path V_WMMA_F32_16X16X4_F32 everywhere.
//
// Round-2 changes:
//  * Weight panels stored TRANSPOSED in LDS ([n][k], stride 34): each WMMA
//    B-fragment is one contiguous ds_load_b64 (no VGPR re-packing movs).
//  * All LDS staging uses GLOBAL_LOAD_ASYNC_TO_LDS_{B32,B128} (ASYNCcnt) --
//    per-lane b32 asyncs perform the transpose scatter; b128 for the
//    contiguous wkv kvA/time_w blocks.
//  * GEMMs double-buffer the LDS panels: DMA for chunk k+1 is issued before
//    computing chunk k (s_wait_asynccnt + barrier after compute).
// ---------------------------------------------------------------------------

#define B_     4
#define T_     1024
#define HID_   1024
#define NATTN_ 1024
#define NHEAD_ 64
#define HS_    16
#define CTX_   1024

#define KC_    32            // k-chunk staged in LDS per GEMM iteration
#define KSTR_  34            // transposed panel k-stride (floats), 8B-aligned
#define PANEL_ (64 * KSTR_)  // one 64(n) x 32(k) panel, padded

typedef __attribute__((ext_vector_type(2))) float v2f;
typedef __attribute__((ext_vector_type(8))) float v8f;

__device__ __forceinline__ v8f vzero8() {
  v8f v = {0.f, 0.f, 0.f, 0.f, 0.f, 0.f, 0.f, 0.f};
  return v;
}

// 16x16x4 fp32 WMMA: D = A(16x4) * B(4x16) + C(16x16), fp32 exact.
__device__ __forceinline__ v8f wmma_f32(v2f a, v2f b, v8f c) {
  return __builtin_amdgcn_wmma_f32_16x16x4_f32(
      /*neg_a=*/false, a, /*neg_b=*/false, b,
      /*c_mod=*/(short)0, c, /*reuse_a=*/false, /*reuse_b=*/false);
}

// ---- async global->LDS DMA (ASYNCcnt-tracked, per-lane addressing) --------
// Generic LDS pointer's low 32 bits are the wave-relative LDS byte offset.
__device__ __forceinline__ void async_copy_b32(void* lds_dst, const void* gsrc) {
  asm volatile("global_load_async_to_lds_b32 %0, %1, off"
               :: "v"((unsigned)(unsigned long long)lds_dst), "v"(gsrc)
               : "memory");
}
__device__ __forceinline__ void async_copy_b128(void* lds_dst, const void* gsrc) {
  asm volatile("global_load_async_to_lds_b128 %0, %1, off"
               :: "v"((unsigned)(unsigned long long)lds_dst), "v"(gsrc)
               : "memory");
}
__device__ __forceinline__ void wait_async() {
  asm volatile("s_wait_asynccnt 0" ::: "memory");
}

// ---------------------------------------------------------------------------
// Kernel 1: fused K/V/R GEMM + epilogue.
// Grid: 32 m-blocks (128 rows) x 16 n-groups = 512 blocks, 8 waves.
// Each wave: one 16(M) x 64(N) tile of all three outputs (12 v8f accums).
// Double-buffered async-staged transposed weight panels in LDS.
// Fragment layout (ISA 7.12.2, 32-bit):
//   A(m,k): m = lane&15, k = j + 2*(lane>>4)
//   B(k,n): n = lane&15, k = j + 2*(lane>>4)   -> contiguous in [n][k] LDS
//   D(m,n): n = lane&15, m = j + 8*(lane>>4)
// ---------------------------------------------------------------------------
__global__ __launch_bounds__(256)
void kvr_gemm_kernel(const float* __restrict__ x,
                     const float* __restrict__ Wk, const float* __restrict__ bk,
                     const float* __restrict__ Wv, const float* __restrict__ bv,
                     const float* __restrict__ Wr, const float* __restrict__ br,
                     const float* __restrict__ alpha,   // (H,1,CTX)
                     float* __restrict__ kexp, float* __restrict__ kvA,
                     float* __restrict__ rsig) {
  __shared__ float ldsK[2 * PANEL_];
  __shared__ float ldsV[2 * PANEL_];
  __shared__ float ldsR[2 * PANEL_];

  const int lane = threadIdx.x & 31;
  const int wave = threadIdx.x >> 5;
  const int mblock = blockIdx.x >> 4;     // 0..31 (128 M-rows each)
  const int ngrp   = blockIdx.x & 15;     // 0..15 (64 N-cols each)
  const int mtile  = mblock * 8 + wave;   // 0..255
  const int m16 = mtile * 16;
  const int n64 = ngrp * 64;
  const int l15 = lane & 15;
  const int lh  = lane >> 4;

  const int b  = m16 / T_;                // blocks never straddle a batch
  const int t0 = m16 % T_;
  const int mrow = t0 + l15;              // this lane's A-row (time index)
  // branch-free time shift: clamped previous row, zeroed via scale at t==0
  const int   mprev = (mrow > 0) ? (mrow - 1) : 0;
  const float sprev = (mrow > 0) ? 1.f : 0.f;
  const float* __restrict__ xrowCur  = x + (size_t)(b * T_ + mrow)  * HID_;
  const float* __restrict__ xrowPrev = x + (size_t)(b * T_ + mprev) * HID_;

  v8f accK[4], accV[4], accR[4];
#pragma unroll
  for (int i = 0; i < 4; ++i) { accK[i] = vzero8(); accV[i] = vzero8(); accR[i] = vzero8(); }

  // async transpose-scatter staging of one 32(k) x 64(n) panel per weight
  auto stage = [&](int kc, int slot) {
#pragma unroll
    for (int r = 0; r < 8; ++r) {                 // 2048 elems / 256 threads
      const int i  = threadIdx.x + r * 256;
      const int kk = i >> 6;                      // 0..31
      const int nn = i & 63;                      // 0..63
      const size_t g = (size_t)(kc + kk) * NATTN_ + n64 + nn;
      const int   d = slot * PANEL_ + nn * KSTR_ + kk;
      async_copy_b32(&ldsK[d], &Wk[g]);
      async_copy_b32(&ldsV[d], &Wv[g]);
      async_copy_b32(&ldsR[d], &Wr[g]);
    }
  };

  stage(0, 0);
  wait_async();
  __syncthreads();

  int cur = 0;
  for (int kc = 0; kc < HID_; kc += KC_) {
    if (kc + KC_ < HID_) stage(kc + KC_, cur ^ 1);  // prefetch next chunk

    // chunk-uniform time-shift select (32-k chunks never straddle 512)
    const float* __restrict__ xrow = (kc < HID_ / 2) ? xrowPrev : xrowCur;
    const float aScale = (kc < HID_ / 2) ? sprev : 1.f;
    const float* __restrict__ pK = ldsK + cur * PANEL_;
    const float* __restrict__ pV = ldsV + cur * PANEL_;
    const float* __restrict__ pR = ldsR + cur * PANEL_;

    for (int k0 = 0; k0 < KC_; k0 += 4) {
      const int kbL = k0 + lh * 2;              // k = j + 2*(lane>>4)
      v2f a = *(const v2f*)(xrow + kc + kbL);   // b64 load, branch-free
      a.x *= aScale;
      a.y *= aScale;
#pragma unroll
      for (int nt = 0; nt < 4; ++nt) {
        const int c0 = (nt * 16 + l15) * KSTR_ + kbL;
        const v2f fK = *(const v2f*)&pK[c0];    // single ds_load_b64 each
        const v2f fV = *(const v2f*)&pV[c0];
        const v2f fR = *(const v2f*)&pR[c0];
        accK[nt] = wmma_f32(a, fK, accK[nt]);
        accV[nt] = wmma_f32(a, fV, accV[nt]);
        accR[nt] = wmma_f32(a, fR, accR[nt]);
      }
    }
    wait_async();       // next chunk's DMA has landed
    __syncthreads();    // everyone done reading `cur`, next data visible
    cur ^= 1;
  }

  // ---- epilogue: bias, exp-clip, fold alpha into kv, sigmoid(r) ----
#pragma unroll
  for (int nt = 0; nt < 4; ++nt) {
    const int n = n64 + nt * 16 + l15;
    const int h = n >> 4;
    const float bkv = bk[n], bvv = bv[n], brv = br[n];
#pragma unroll
    for (int j = 0; j < 8; ++j) {
      const int t = t0 + j + lh * 8;            // D: m = j + 8*(lane>>4)
      const size_t idx = (size_t)(b * T_ + t) * NATTN_ + n;
      float kval = accK[nt][j] + bkv;
      kval = fminf(fmaxf(kval, -60.f), 30.f);
      const float ke = expf(kval);
      const float vv = accV[nt][j] + bvv;
      const float rv = accR[nt][j] + brv;
      kexp[idx] = ke;
      kvA[idx]  = ke * vv * alpha[(size_t)h * CTX_ + t];
      rsig[idx] = 1.f / (1.f + expf(-rv));
    }
  }
}

// ---------------------------------------------------------------------------
// Kernel 2: cumulative sum of kexp over time per (b, channel) column.
// ---------------------------------------------------------------------------
__global__ __launch_bounds__(256)
void cumsum_kernel(const float* __restrict__ kexp, float* __restrict__ sumk) {
  const int col = blockIdx.x * 256 + threadIdx.x;   // 0..4095
  const int b = col >> 10;
  const int n = col & 1023;
  float s = 0.f;
  for (int t = 0; t < T_; ++t) {
    const size_t idx = (size_t)(b * T_ + t) * NATTN_ + n;
    s += kexp[idx];
    sumk[idx] = s;
  }
}

// ---------------------------------------------------------------------------
// Kernel 3: causal Toeplitz matmul per (b,h):
//   wkv[t,c] = sum_{u<=t} time_w[h, 1023-t+u] * kvA[u,c]
// Entire kvA block (64KB) + time_w row (4KB) async-DMA'd into LDS once
// (b128, contiguous rows). Unmasked full tiles split from the masked
// diagonal tile. Epilogue fuses beta[h,t], sigmoid(r), /sum_k.
// ---------------------------------------------------------------------------
__global__ __launch_bounds__(256)
void wkv_kernel(const float* __restrict__ time_w,   // (H, CTX)
                const float* __restrict__ beta,     // (H, CTX, 1)
                const float* __restrict__ kvA,
                const float* __restrict__ rsig,
                const float* __restrict__ sumk,
                float* __restrict__ rwkv) {
  extern __shared__ float smem[];
  float* ldsKV = smem;               // [1024][16]
  float* ldsTW = smem + T_ * HS_;    // [1024]

  const int bh = blockIdx.x;         // 0..255
  const int b  = bh >> 6;
  const int h  = bh & 63;
  const int lane = threadIdx.x & 31;
  const int wave = threadIdx.x >> 5;
  const int l15 = lane & 15;
  const int lh  = lane >> 4;

  // ---- async staging: kvA column block + time_w row (b128 DMA) ----
  const float* __restrict__ srcKV = kvA + (size_t)(b * T_) * NATTN_ + h * HS_;
  for (int i = threadIdx.x; i < T_ * 4; i += 256) {   // 4 xfloat4 per 16-row
    const int u = i >> 2, q = i & 3;
    async_copy_b128(&ldsKV[u * HS_ + q * 4], srcKV + (size_t)u * NATTN_ + q * 4);
  }
  const float* __restrict__ tw = time_w + (size_t)h * CTX_;
  {
    const int i = threadIdx.x;                        // 256 float4 = 1024 f
    async_copy_b128(&ldsTW[i * 4], tw + i * 4);
  }
  wait_async();
  __syncthreads();

  for (int ti = wave; ti < T_ / 16; ti += 8) {
    v8f acc = vzero8();
    const int tA = ti * 16 + l15;               // A-row time (m = lane&15)

    // full (unmasked) u-tiles: u < ti*16 <= tA always
    for (int ui = 0; ui < ti; ++ui) {
#pragma unroll
      for (int q = 0; q < 4; ++q) {
        const int u0 = ui * 16 + q * 4 + lh * 2;
        v2f a, fB;
        a.x = ldsTW[CTX_ - 1 - tA + u0];
        a.y = ldsTW[CTX_ - 1 - tA + u0 + 1];
        fB.x = ldsKV[u0 * HS_ + l15];
        fB.y = ldsKV[(u0 + 1) * HS_ + l15];
        acc = wmma_f32(a, fB, acc);
      }
    }
    // diagonal tile: branch-free causal mask (wrapped index + select)
    {
      const int ui = ti;
#pragma unroll
      for (int q = 0; q < 4; ++q) {
        const int u0 = ui * 16 + q * 4 + lh * 2;
        const int u1 = u0 + 1;
        v2f a, fB;
        a.x = (u0 <= tA) ? ldsTW[(CTX_ - 1 - tA + u0) & (CTX_ - 1)] : 0.f;
        a.y = (u1 <= tA) ? ldsTW[(CTX_ - 1 - tA + u1) & (CTX_ - 1)] : 0.f;
        fB.x = ldsKV[u0 * HS_ + l15];
        fB.y = ldsKV[u1 * HS_ + l15];
        acc = wmma_f32(a, fB, acc);
      }
    }
    // epilogue: beta * acc * sigmoid(r) / sum_k
#pragma unroll
    for (int j = 0; j < 8; ++j) {
      const int t = ti * 16 + j + lh * 8;
      const int n = h * HS_ + l15;
      const size_t idx = (size_t)(b * T_ + t) * NATTN_ + n;
      const float val = acc[j] * beta[(size_t)h * CTX_ + t];
      rwkv[idx] = rsig[idx] * val / sumk[idx];
    }
  }
}

// ---------------------------------------------------------------------------
// Kernel 4: output GEMM  out = (rwkv @ Wo + bo) * gamma[t]
// Same double-buffered async-staged transposed tiling as kernel 1.
// ---------------------------------------------------------------------------
__global__ __launch_bounds__(256)
void out_gemm_kernel(const float* __restrict__ rwkv,
                     const float* __restrict__ Wo, const float* __restrict__ bo,
                     const float* __restrict__ gamma,   // (CTX,1)
                     float* __restrict__ out) {
  __shared__ float ldsO[2 * PANEL_];

  const int lane = threadIdx.x & 31;
  const int wave = threadIdx.x >> 5;
  const int mblock = blockIdx.x >> 4;
  const int ngrp   = blockIdx.x & 15;
  const int mtile  = mblock * 8 + wave;
  const int m16 = mtile * 16;
  const int n64 = ngrp * 64;
  const int l15 = lane & 15;
  const int lh  = lane >> 4;

  const int b  = m16 / T_;
  const int t0 = m16 % T_;
  const int mrow = t0 + l15;
  const float* __restrict__ arow = rwkv + (size_t)(b * T_ + mrow) * NATTN_;

  v8f acc[4];
#pragma unroll
  for (int i = 0; i < 4; ++i) acc[i] = vzero8();

  auto stage = [&](int kc, int slot) {
#pragma unroll
    for (int r = 0; r < 8; ++r) {
      const int i  = threadIdx.x + r * 256;
      const int kk = i >> 6;
      const int nn = i & 63;
      async_copy_b32(&ldsO[slot * PANEL_ + nn * KSTR_ + kk],
                     &Wo[(size_t)(kc + kk) * HID_ + n64 + nn]);
    }
  };

  stage(0, 0);
  wait_async();
  __syncthreads();

  int cur = 0;
  for (int kc = 0; kc < NATTN_; kc += KC_) {
    if (kc + KC_ < NATTN_) stage(kc + KC_, cur ^ 1);
    const float* __restrict__ pO = ldsO + cur * PANEL_;

    for (int k0 = 0; k0 < KC_; k0 += 4) {
      const int kbL = k0 + lh * 2;
      const v2f a = *(const v2f*)(arow + kc + kbL);
#pragma unroll
      for (int nt = 0; nt < 4; ++nt) {
        const v2f fB = *(const v2f*)&pO[(nt * 16 + l15) * KSTR_ + kbL];
        acc[nt] = wmma_f32(a, fB, acc[nt]);
      }
    }
    wait_async();
    __syncthreads();
    cur ^= 1;
  }

#pragma unroll
  for (int nt = 0; nt < 4; ++nt) {
    const int n = n64 + nt * 16 + l15;
    const float bon = bo[n];
#pragma unroll
    for (int j = 0; j < 8; ++j) {
      const int t = t0 + j + lh * 8;
      const size_t idx = (size_t)(b * T_ + t) * HID_ + n;
      out[idx] = (acc[nt][j] + bon) * gamma[t];
    }
  }
}

// ---------------------------------------------------------------------------
// Host launcher. Input order: x, time_w, time_alpha, time_beta, time_gamma,
// Wk, bk, Wv, bv, Wr, br, Wo, bo. All fp32. Output fp32.
// Workspace: 5 buffers of B*T*N_ATTN floats (16MB each, 80MB total).
// ---------------------------------------------------------------------------
extern "C" void kernel_launch(void* const* d_in, const int* in_sizes, int n_in,
                              void* d_out, int out_size, void* d_ws, size_t ws_size,
                              hipStream_t stream) {
  (void)in_sizes; (void)n_in; (void)out_size; (void)ws_size;

  const float* x       = (const float*)d_in[0];
  const float* time_w  = (const float*)d_in[1];
  const float* alpha   = (const float*)d_in[2];
  const float* beta    = (const float*)d_in[3];
  const float* gamma   = (const float*)d_in[4];
  const float* Wk      = (const float*)d_in[5];
  const float* bk      = (const float*)d_in[6];
  const float* Wv      = (const float*)d_in[7];
  const float* bv      = (const float*)d_in[8];
  const float* Wr      = (const float*)d_in[9];
  const float* br      = (const float*)d_in[10];
  const float* Wo      = (const float*)d_in[11];
  const float* bo      = (const float*)d_in[12];
  float* out = (float*)d_out;

  const size_t NELEM = (size_t)B_ * T_ * NATTN_;   // 4M floats
  float* ws   = (float*)d_ws;
  float* kexp = ws + 0 * NELEM;
  float* kvA  = ws + 1 * NELEM;
  float* rsig = ws + 2 * NELEM;
  float* sumk = ws + 3 * NELEM;
  float* rwkv = ws + 4 * NELEM;

  // 1) fused k/v/r GEMM: 32 m-blocks x 16 n-groups
  kvr_gemm_kernel<<<512, 256, 0, stream>>>(x, Wk, bk, Wv, bv, Wr, br, alpha,
                                           kexp, kvA, rsig);
  // 2) cumsum over time: 4096 columns
  cumsum_kernel<<<16, 256, 0, stream>>>(kexp, sumk);
  // 3) per-(b,h) causal Toeplitz matmul (LDS-resident kvA + time_w)
  const size_t wkv_smem = (size_t)(T_ * HS_ + CTX_) * sizeof(float); // 68KB
  wkv_kernel<<<B_ * NHEAD_, 256, wkv_smem, stream>>>(time_w, beta, kvA, rsig,
                                                     sumk, rwkv);
  // 4) output GEMM with bias + gamma epilogue
  out_gemm_kernel<<<512, 256, 0, stream>>>(rwkv, Wo, bo, gamma, out);
}